// CPC_20418274525429
// MI455X (gfx1250) — compile-verified
//
#include <hip/hip_runtime.h>
#include <math.h>

typedef __attribute__((ext_vector_type(2))) float v2f;
typedef __attribute__((ext_vector_type(8))) float v8f;

#define S_DIM      512
#define N_DIM      1024
#define H_DIM      128
#define Z_DIM      64
#define START_T    64          // S//8
#define END_T      417         // S - 8 - 85 - 4 + 2
#define CNT_T      353         // END_T - START_T
#define NEG_DIST   85          // S//6
#define TSPAN      4
#define NSAMP      8

// ---- workspace layout (in floats) ----
#define WS_H_MEAN   0
#define WS_Z_MEAN   (WS_H_MEAN + S_DIM*H_DIM)     // after 65536
#define WS_H_POOL   (WS_Z_MEAN + S_DIM*Z_DIM)
#define WS_Z_POOL   (WS_H_POOL + S_DIM*H_DIM)
#define WS_C_PHI    (WS_Z_POOL + S_DIM*H_DIM)
#define WS_ZNORM    (WS_C_PHI  + S_DIM*H_DIM)
#define WS_CNORM    (WS_ZNORM  + S_DIM)
#define WS_COLSUM   (WS_CNORM  + S_DIM)
#define WS_PARTIAL  (WS_COLSUM + H_DIM)

// ------------------------------------------------------------------
// 1) Streaming mean over nodes (the bandwidth-bound 402 MB pass).
//    blocks [0,512): all_h -> h_mean[s,128]; blocks [512,1024): all_z -> z_mean[s,64]
//    float4 (b128) loads, fixed-order accumulation, LDS combine.
// ------------------------------------------------------------------
__global__ void cpc_mean_kernel(const float* __restrict__ all_h,
                                const float* __restrict__ all_z,
                                float* __restrict__ h_mean,
                                float* __restrict__ z_mean) {
  __shared__ float4 sm[256];
  const int b = blockIdx.x;
  const float4* src;
  float* dst;
  int FQ;                                   // float4s per row
  if (b < S_DIM) {
    src = (const float4*)(all_h + (size_t)b * N_DIM * H_DIM);
    dst = h_mean + (size_t)b * H_DIM;
    FQ = H_DIM / 4;                         // 32
  } else {
    src = (const float4*)(all_z + (size_t)(b - S_DIM) * N_DIM * Z_DIM);
    dst = z_mean + (size_t)(b - S_DIM) * Z_DIM;
    FQ = Z_DIM / 4;                         // 16
  }
  const int G = 256 / FQ;
  const int q = threadIdx.x % FQ;
  const int g = threadIdx.x / FQ;
  float4 acc = make_float4(0.f, 0.f, 0.f, 0.f);
  for (int n = g; n < N_DIM; n += G) {
    float4 v = src[(size_t)n * FQ + q];
    acc.x += v.x; acc.y += v.y; acc.z += v.z; acc.w += v.w;
  }
  sm[threadIdx.x] = acc;
  __syncthreads();
  if ((int)threadIdx.x < FQ) {
    float4 s = make_float4(0.f, 0.f, 0.f, 0.f);
    for (int gg = 0; gg < G; ++gg) {
      float4 v = sm[gg * FQ + q];
      s.x += v.x; s.y += v.y; s.z += v.z; s.w += v.w;
    }
    const float inv = 1.0f / (float)N_DIM;
    dst[q * 4 + 0] = s.x * inv;
    dst[q * 4 + 1] = s.y * inv;
    dst[q * 4 + 2] = s.z * inv;
    dst[q * 4 + 3] = s.w * inv;
  }
}

// ------------------------------------------------------------------
// 2) out[M,N] = A[M,K] @ W[N,K]^T + bias[N]  via V_WMMA_F32_16X16X4_F32
//    One wave per 16x16 output tile; K-loop in steps of 4.
//    A frag: lane reads A[tm*16 + lane%16, 4k + 2*(lane>>4) + {0,1}]
//    B frag (= W^T): identical addressing pattern on W rows.
//    C/D: vgpr r <-> row r + 8*(lane>>4), col lane%16.
// ------------------------------------------------------------------
__global__ void cpc_gemm_wmma(const float* __restrict__ A,
                              const float* __restrict__ W,
                              const float* __restrict__ bias,
                              float* __restrict__ out,
                              int M, int N, int K) {
  const int nt   = N >> 4;
  const int tile = blockIdx.x;
  const int tm   = tile / nt;
  const int tn   = tile % nt;
  const int lane = threadIdx.x;     // 0..31, all active (EXEC must be ~0)
  const int r    = lane & 15;
  const int half = lane >> 4;

  const float* arow = A + (size_t)(tm * 16 + r) * K + 2 * half;
  const float* wrow = W + (size_t)(tn * 16 + r) * K + 2 * half;

  v8f c;
  const float bv = bias[tn * 16 + r];       // bias depends only on column
  for (int i = 0; i < 8; ++i) c[i] = bv;

  for (int kb = 0; kb < K; kb += 4) {
    v2f a = *(const v2f*)(arow + kb);
    v2f w = *(const v2f*)(wrow + kb);
    c = __builtin_amdgcn_wmma_f32_16x16x4_f32(
        /*neg_a=*/false, a, /*neg_b=*/false, w,
        /*c_mod=*/(short)0, c, /*reuse_a=*/false, /*reuse_b=*/false);
  }

  float* obase = out + (size_t)(tm * 16 + 8 * half) * N + tn * 16 + r;
  for (int i = 0; i < 8; ++i) obase[(size_t)i * N] = c[i];
}

// ------------------------------------------------------------------
// 3) Row L2 norms for z_pool (blocks 0..511) and c_phi (blocks 512..1023)
// ------------------------------------------------------------------
__global__ void cpc_norms_kernel(const float* __restrict__ z_pool,
                                 const float* __restrict__ c_phi,
                                 float* __restrict__ znorm,
                                 float* __restrict__ cnorm) {
  __shared__ float sm[128];
  const int b = blockIdx.x;
  const float* row;
  float* out;
  int r;
  if (b < S_DIM) { r = b;         row = z_pool + (size_t)r * H_DIM; out = znorm; }
  else           { r = b - S_DIM; row = c_phi  + (size_t)r * H_DIM; out = cnorm; }
  float v = row[threadIdx.x];
  sm[threadIdx.x] = v * v;
  __syncthreads();
  for (int s = 64; s > 0; s >>= 1) {
    if ((int)threadIdx.x < s) sm[threadIdx.x] += sm[threadIdx.x + s];
    __syncthreads();
  }
  if (threadIdx.x == 0) out[r] = sqrtf(sm[0]);
}

// ------------------------------------------------------------------
// 4) Column sums of z_pool (for global mean / distance term)
// ------------------------------------------------------------------
__global__ void cpc_colsum_kernel(const float* __restrict__ z_pool,
                                  float* __restrict__ colsum) {
  const int f = threadIdx.x;        // 0..127
  float s = 0.f;
  for (int i = 0; i < S_DIM; ++i) s += z_pool[(size_t)i * H_DIM + f];
  colsum[f] = s;
}

// ------------------------------------------------------------------
// 5) NCE per-t partial: one block per t; 32 (i,m) pairs x 8 threads/dot
// ------------------------------------------------------------------
__global__ void cpc_nce_kernel(const float* __restrict__ z_pool,
                               const float* __restrict__ c_phi,
                               const float* __restrict__ znorm,
                               const float* __restrict__ cnorm,
                               float* __restrict__ partial) {
  __shared__ float red[256];
  __shared__ float sim[32];
  const int t    = START_T + blockIdx.x;
  const int tid  = threadIdx.x;
  const int pair = tid >> 3;        // 0..31
  const int sub  = tid & 7;         // 0..7
  const int i    = (pair >> 3) + 1; // 1..4
  const int m    = pair & 7;        // 0..7
  const int off  = (m == 0) ? 0 : (NEG_DIST + m - 1);
  const int idx  = t + i + off;     // max 416+4+91 = 511, in range

  const float* enc = z_pool + (size_t)idx * H_DIM;
  const float* ctx = c_phi  + (size_t)t   * H_DIM;

  float acc = 0.f;
  const int k0 = sub * 16;
  for (int k = k0; k < k0 + 16; ++k) acc += enc[k] * ctx[k];
  red[tid] = acc;
  __syncthreads();

  if (sub == 0) {
    float s = 0.f;
    for (int u = 0; u < 8; ++u) s += red[pair * 8 + u];
    const float denom = fmaxf(znorm[idx], 1e-8f) * fmaxf(cnorm[t], 1e-8f);
    sim[pair] = s / denom;
  }
  __syncthreads();

  if (tid < 4) {                    // one thread per i: log-softmax over m, take m==0
    float mx = sim[tid * 8];
    for (int u = 1; u < 8; ++u) mx = fmaxf(mx, sim[tid * 8 + u]);
    float se = 0.f;
    for (int u = 0; u < 8; ++u) se += expf(sim[tid * 8 + u] - mx);
    red[tid] = sim[tid * 8] - (mx + logf(se));
  }
  __syncthreads();
  if (tid == 0) partial[blockIdx.x] = red[0] + red[1] + red[2] + red[3];
}

// ------------------------------------------------------------------
// 6) Final NCE reduce -> d_out[0]
// ------------------------------------------------------------------
__global__ void cpc_nce_final_kernel(const float* __restrict__ partial,
                                     float* __restrict__ d_out) {
  __shared__ float sm[256];
  const int tid = threadIdx.x;
  float a = 0.f;
  for (int e = tid; e < CNT_T; e += 256) a += partial[e];
  sm[tid] = a;
  __syncthreads();
  for (int s = 128; s > 0; s >>= 1) {
    if (tid < s) sm[tid] += sm[tid + s];
    __syncthreads();
  }
  if (tid == 0) d_out[0] = sm[0] / (-1.0f * (float)CNT_T * (float)TSPAN);
}

// ------------------------------------------------------------------
// 7) Distance term -> d_out[1]
// ------------------------------------------------------------------
__global__ void cpc_distance_kernel(const float* __restrict__ z_pool,
                                    const float* __restrict__ colsum,
                                    float* __restrict__ d_out) {
  __shared__ float sm[256];
  const int tid = threadIdx.x;
  const float invS = 1.0f / (float)S_DIM;
  float acc = 0.f;
  for (int e = tid; e < S_DIM * H_DIM; e += 256) {
    const int f = e & (H_DIM - 1);
    const float d = z_pool[e] - colsum[f] * invS;
    acc += d * d;
  }
  sm[tid] = acc;
  __syncthreads();
  for (int s = 128; s > 0; s >>= 1) {
    if (tid < s) sm[tid] += sm[tid + s];
    __syncthreads();
  }
  if (tid == 0) d_out[1] = sm[0] * invS;
}

// ------------------------------------------------------------------
extern "C" void kernel_launch(void* const* d_in, const int* in_sizes, int n_in,
                              void* d_out, int out_size, void* d_ws, size_t ws_size,
                              hipStream_t stream) {
  const float* all_h = (const float*)d_in[0];
  const float* all_z = (const float*)d_in[1];
  const float* Wh    = (const float*)d_in[2];
  const float* bh    = (const float*)d_in[3];
  const float* Wz    = (const float*)d_in[4];
  const float* bz    = (const float*)d_in[5];
  const float* Wphi  = (const float*)d_in[6];
  const float* bphi  = (const float*)d_in[7];

  float* ws      = (float*)d_ws;
  float* h_mean  = ws + WS_H_MEAN;
  float* z_mean  = ws + WS_Z_MEAN;
  float* h_pool  = ws + WS_H_POOL;
  float* z_pool  = ws + WS_Z_POOL;
  float* c_phi   = ws + WS_C_PHI;
  float* znorm   = ws + WS_ZNORM;
  float* cnorm   = ws + WS_CNORM;
  float* colsum  = ws + WS_COLSUM;
  float* partial = ws + WS_PARTIAL;
  float* out     = (float*)d_out;

  // 1) streaming means (dominant cost; HBM-bound)
  cpc_mean_kernel<<<2 * S_DIM, 256, 0, stream>>>(all_h, all_z, h_mean, z_mean);

  // 2) projections: one wave per 16x16 tile, (512/16)*(128/16) = 256 tiles
  const int tiles = (S_DIM / 16) * (H_DIM / 16);
  cpc_gemm_wmma<<<tiles, 32, 0, stream>>>(h_mean, Wh,   bh,   h_pool, S_DIM, H_DIM, H_DIM);
  cpc_gemm_wmma<<<tiles, 32, 0, stream>>>(z_mean, Wz,   bz,   z_pool, S_DIM, H_DIM, Z_DIM);
  cpc_gemm_wmma<<<tiles, 32, 0, stream>>>(h_pool, Wphi, bphi, c_phi,  S_DIM, H_DIM, H_DIM);

  // 3) norms + column sums
  cpc_norms_kernel<<<2 * S_DIM, 128, 0, stream>>>(z_pool, c_phi, znorm, cnorm);
  cpc_colsum_kernel<<<1, H_DIM, 0, stream>>>(z_pool, colsum);

  // 4) NCE + final scalars
  cpc_nce_kernel<<<CNT_T, 256, 0, stream>>>(z_pool, c_phi, znorm, cnorm, partial);
  cpc_nce_final_kernel<<<1, 256, 0, stream>>>(partial, out);
  cpc_distance_kernel<<<1, 256, 0, stream>>>(z_pool, colsum, out);
}